// IRadon_31507880083523
// MI455X (gfx1250) — compile-verified
//
#include <hip/hip_runtime.h>
#include <math.h>

// Problem constants (from reference): x [B=2, C=1, D=512, A=180] -> reco [2,1,512,512]
#define BB   2
#define DD   512
#define AA   180
#define SS   512
#define PI_F 3.14159265358979323846f

typedef float v2f __attribute__((ext_vector_type(2)));
typedef float v8f __attribute__((ext_vector_type(8)));

// Analytic ramp-filter spatial kernel (circular conv kernel = 2*f), branchless:
//   delta == 0        -> 0.5
//   |delta| odd       -> -2/(pi^2) * (1/k^2)
//   |delta| even != 0 -> 0
__device__ __forceinline__ float gval(int delta) {
    const int   ad = delta < 0 ? -delta : delta;
    const float fk = (float)ad;
    const float r  = __builtin_amdgcn_rcpf(fk * fk);          // v_rcp_f32
    float v = (ad & 1) ? (-2.0f / (PI_F * PI_F)) * r : 0.0f;  // v_cndmask
    v = (ad == 0) ? 0.5f : v;                                 // v_cndmask
    return v;
}

// -------- Kernel 1: ramp filtering as GEMM  Y_b = G(512x512) * X_b(512x180) --------
// One wave (32 threads) per 16x16 output tile, fp32 WMMA 16x16x4.
// Grid: (M/16=32, ceil(180/16)=12, B=2). A-matrix (G) generated analytically in-register.
__global__ __launch_bounds__(32) void ramp_filter_wmma(const float* __restrict__ x,
                                                       float* __restrict__ xf) {
    const int tileM = blockIdx.x;           // 0..31  (detector-bin tile)
    const int tileN = blockIdx.y;           // 0..11  (angle tile)
    const int b     = blockIdx.z;           // batch
    const int lane  = threadIdx.x;          // 0..31
    const int half  = lane >> 4;            // 0: lanes 0-15, 1: lanes 16-31
    const int l15   = lane & 15;

    const int mrow = tileM * 16 + l15;      // output detector bin (A-matrix row)
    const int ncol = tileN * 16 + l15;      // angle column (B/D-matrix col)
    const int nc   = ncol < (AA - 1) ? ncol : (AA - 1);   // clamped (always-valid) column
    const float nmask = (ncol < AA) ? 1.0f : 0.0f;        // zero out OOB columns

    const float* __restrict__ X = x + (size_t)b * DD * AA + nc;

    v8f acc = {};
    for (int k = 0; k < DD; k += 4) {
        const int kk = k + half * 2;        // this lane's K pair

        // A fragment: G[mrow][kk], G[mrow][kk+1] (analytic, branchless, no memory traffic)
        v2f afrag;
        afrag.x = gval(mrow - kk);
        afrag.y = gval(mrow - (kk + 1));

        // B fragment: X[kk][ncol], X[kk+1][ncol] (unconditional loads, masked by v_cndmask-able mul)
        v2f bfrag;
        bfrag.x = X[kk * AA] * nmask;
        bfrag.y = X[(kk + 1) * AA] * nmask;

        // D = A*B + C : v_wmma_f32_16x16x4_f32
        acc = __builtin_amdgcn_wmma_f32_16x16x4_f32(
            /*neg_a=*/false, afrag, /*neg_b=*/false, bfrag,
            /*c_mod=*/(short)0, acc, /*reuse_a=*/false, /*reuse_b=*/false);
    }

    // C/D layout: lane holds N=ncol; VGPR r -> M = r + 8*half
    if (ncol < AA) {
        float* __restrict__ Y = xf + (size_t)b * DD * AA + ncol;
        #pragma unroll
        for (int r = 0; r < 8; ++r) {
            const int m = tileM * 16 + r + half * 8;
            Y[m * AA] = acc[r];
        }
    }
}

// -------- Kernel 2: backprojection --------
// One thread per output pixel; cos/sin tables staged in LDS once per block.
// Gather is branchless: clip index, multiply by validity (matches reference semantics).
__global__ __launch_bounds__(256) void backproject(const float* __restrict__ xf,
                                                   float* __restrict__ out) {
    __shared__ float s_cos[AA];
    __shared__ float s_sin[AA];

    const int t = threadIdx.x;
    if (t < AA) {
        const float th = (float)t * (PI_F / 180.0f);  // deg2rad(a)
        s_cos[t] = cosf(th);
        s_sin[t] = sinf(th);
    }
    __syncthreads();

    const int gid = blockIdx.x * 256 + t;   // 0 .. 2*512*512-1
    const int j = gid & (SS - 1);           // x index
    const int i = (gid >> 9) & (SS - 1);    // y index
    const int b = gid >> 18;                // batch

    const float xg = (float)j * (2.0f / (float)(SS - 1)) - 1.0f;
    const float yg = (float)i * (2.0f / (float)(SS - 1)) - 1.0f;

    const float* __restrict__ Xf = xf + (size_t)b * DD * AA;

    float acc = 0.0f;
    for (int a = 0; a < AA; ++a) {
        const float T   = xg * s_cos[a] - yg * s_sin[a];
        const float pix = (T + 1.0f) * ((float)(DD - 1) * 0.5f);
        const float fi  = floorf(pix);
        const float w   = pix - fi;
        const int i0 = (int)fi;
        const int i1 = i0 + 1;

        // Clip to valid range (loads always in-bounds), mask contributions.
        const int c0 = i0 < 0 ? 0 : (i0 > DD - 1 ? DD - 1 : i0);
        const int c1 = i1 < 0 ? 0 : (i1 > DD - 1 ? DD - 1 : i1);
        const float m0 = (i0 >= 0 && i0 <= DD - 1) ? 1.0f : 0.0f;
        const float m1 = (i1 >= 0 && i1 <= DD - 1) ? 1.0f : 0.0f;

        const float v0 = Xf[c0 * AA + a] * m0;
        const float v1 = Xf[c1 * AA + a] * m1;
        acc += v0 * (1.0f - w) + v1 * w;
    }

    const float mask = (xg * xg + yg * yg <= 1.0f) ? 1.0f : 0.0f;
    out[gid] = acc * mask * (PI_F / (2.0f * (float)AA));
}

extern "C" void kernel_launch(void* const* d_in, const int* in_sizes, int n_in,
                              void* d_out, int out_size, void* d_ws, size_t ws_size,
                              hipStream_t stream) {
    (void)in_sizes; (void)n_in; (void)out_size; (void)ws_size;

    const float* x   = (const float*)d_in[0];   // [2,1,512,180] f32
    float*       out = (float*)d_out;           // [2,1,512,512] f32
    float*       xf  = (float*)d_ws;            // filtered sinogram scratch: 2*512*180 f32

    // 1) Ramp filter via fp32 WMMA GEMM: 16x16 tiles, 1 wave each.
    dim3 gGemm(DD / 16, (AA + 15) / 16, BB);    // (32, 12, 2)
    ramp_filter_wmma<<<gGemm, dim3(32), 0, stream>>>(x, xf);

    // 2) Backprojection: one thread per output pixel.
    const int total = BB * SS * SS;             // 524288
    backproject<<<total / 256, dim3(256), 0, stream>>>(xf, out);
}